// SelfAttention_28046136443133
// MI455X (gfx1250) — compile-verified
//
#include <hip/hip_runtime.h>
#include <hip/hip_bf16.h>
#include <math.h>

// Problem constants (from reference): B=32, S=1, DIM=4096, HQ=32, HKV=8, HD=128,
// MAXS=2048, START=2047.
constexpr int kB    = 32;
constexpr int kDIM  = 4096;
constexpr int kHQ   = 32;
constexpr int kHKV  = 8;
constexpr int kHD   = 128;
constexpr int kMAXS = 2048;

typedef __attribute__((ext_vector_type(2))) float v2f;
typedef __attribute__((ext_vector_type(8))) float v8f;
typedef __attribute__((ext_vector_type(4))) unsigned int v4u;
typedef __attribute__((ext_vector_type(8))) int v8i_;
typedef __attribute__((ext_vector_type(4))) int v4i_;

// ---------------------------------------------------------------------------
// Tensor Data Mover path selection
// ---------------------------------------------------------------------------
#if defined(__has_builtin)
#if __has_builtin(__builtin_amdgcn_tensor_load_to_lds) && \
    __has_builtin(__builtin_amdgcn_s_wait_tensorcnt)
#define USE_TDM 1
#if __has_include(<hip/amd_detail/amd_gfx1250_TDM.h>)
#define TDM_6ARG 1   // therock-10.0 headers -> 6-arg builtin
#endif
#endif
#endif

#define GEMM_LDS_STRIDE 68  // 64 data dwords + 4 pad dwords per row

#if defined(USE_TDM)
// Issue one TDM descriptor moving a 32-row x 64-fp32 tile (row stride =
// `stride_elems` fp32) from global memory into LDS at `lds_ptr`, inserting
// 4 pad dwords after every 64 data dwords (LDS row stride 68 = conflict-free
// fragment reads). D# layout per CDNA5 ISA 8.3/8.4.
__device__ __forceinline__ void tdm_load_tile32x64(const float* tile_base,
                                                   int stride_elems,
                                                   void* lds_ptr) {
  const unsigned long long ga = (unsigned long long)(uintptr_t)tile_base;
  v4u g0;
  g0[0] = 1u;                                          // count=1 (valid), user mode
  g0[1] = (unsigned)(uintptr_t)lds_ptr;                // lds_addr (bytes)
  g0[2] = (unsigned)(ga & 0xFFFFFFFFull);              // global_addr[31:0]
  g0[3] = (unsigned)((ga >> 32) & 0x1FFFFFFull)        // global_addr[56:32]
          | (2u << 30);                                // type=2 ("image")
  v8i_ g1;
  g1[0] = (2 << 16)        // data_size = 4 bytes
        | (1 << 20)        // pad_enable
        | (5 << 22)        // pad_interval: 1<<5 = 32 x 8B = 64 dwords
        | (3 << 25);       // pad_amount: 4 dwords
  g1[1] = (64 << 16);      // tensor_dim0 = 64 (bits 79:48, low half)
  g1[2] = (32 << 16);      // tensor_dim0 hi = 0; tensor_dim1 = 32 (bits 111:80, low half)
  g1[3] = (64 << 16);      // tensor_dim1 hi = 0; tile_dim0 = 64
  g1[4] = 32;              // tile_dim1 = 32; tile_dim2 = 0
  g1[5] = stride_elems;    // tensor_dim0_stride[31:0]
  g1[6] = 0;               // stride hi, tensor_dim1_stride lo
  g1[7] = 0;
  const v4i_ z4 = {};
#if defined(TDM_6ARG)
  const v8i_ z8 = {};
  __builtin_amdgcn_tensor_load_to_lds(g0, g1, z4, z4, z8, 0);
#else
  __builtin_amdgcn_tensor_load_to_lds(g0, g1, z4, z4, 0);
#endif
}
#endif  // USE_TDM

// ---------------------------------------------------------------------------
// C[M=32][N] = A[32][K] * W[N][K]^T   (fp32, WMMA f32 16x16x4)
// 128 threads = 4 waves; each wave owns one 16x16 C tile (WG tile 32x32).
// K staged through double-buffered LDS in 64-wide chunks. TDM path: wave 0
// drives the Tensor Data Mover, s_wait_tensorcnt pipelines chunk c+1's DMA
// under chunk c's WMMAs. Fallback: batched coalesced float4 loads.
// Fragment layout (ISA 7.12.2): A 16x4, lane l%16 = row, VGPR0/1 hold
// K={0,1} (lanes 0-15) / K={2,3} (lanes 16-31); B 4x16 symmetric.
// ---------------------------------------------------------------------------
__global__ __launch_bounds__(128) void gemm_xwT_kernel(
    const float* __restrict__ A,   // [32][K] row-major
    const float* __restrict__ W,   // [N][K]  row-major
    float* __restrict__ C,         // [32][ldc]
    int N, int K, int ldc)
{
  __shared__ float As[2][32 * GEMM_LDS_STRIDE];
  __shared__ float Bs[2][32 * GEMM_LDS_STRIDE];

  const int tid   = threadIdx.x;
  const int wave  = tid >> 5;          // 0..3
  const int lane  = tid & 31;
  const int mwarp = (wave & 1) * 16;   // wave's M tile: 0 or 16
  const int nwarp = (wave >> 1) * 16;  // wave's N tile: 0 or 16
  const int n0    = blockIdx.x * 32;

  const int lrow  = lane & 15;         // fragment row within tile
  const int khalf = (lane >> 4) * 2;   // K sub-offset: 0 (lanes 0-15) or 2

  const int nchunks = K / 64;
  v8f acc = {};

#if defined(USE_TDM)
  if (wave == 0) {
    tdm_load_tile32x64(A, K, &As[0][0]);
    tdm_load_tile32x64(W + (size_t)n0 * K, K, &Bs[0][0]);
  }
  for (int c = 0; c < nchunks; ++c) {
    const int buf = c & 1;
    if (wave == 0) {
      if (c + 1 < nchunks) {
        tdm_load_tile32x64(A + (c + 1) * 64, K, &As[buf ^ 1][0]);
        tdm_load_tile32x64(W + (size_t)n0 * K + (c + 1) * 64, K, &Bs[buf ^ 1][0]);
        __builtin_amdgcn_s_wait_tensorcnt(2);  // chunk c complete, c+1 in flight
      } else {
        __builtin_amdgcn_s_wait_tensorcnt(0);
      }
    }
    __syncthreads();
#pragma unroll
    for (int kk = 0; kk < 64; kk += 4) {
      const v2f a = *(const v2f*)(&As[buf][(mwarp + lrow) * GEMM_LDS_STRIDE + kk + khalf]);
      const v2f b = *(const v2f*)(&Bs[buf][(nwarp + lrow) * GEMM_LDS_STRIDE + kk + khalf]);
      acc = __builtin_amdgcn_wmma_f32_16x16x4_f32(
          false, a, false, b, (short)0, acc, false, false);
    }
    __syncthreads();
  }
#else
  // Fallback: cooperative loader, thread -> (row tid/16 + 8*i, col 4*(tid%16))
  const int lr = tid >> 4;             // 0..7
  const int lc = (tid & 15) * 4;       // 0..60
  for (int c = 0; c < nchunks; ++c) {
    const int k0 = c * 64;
    float4 ra[4], rw[4];
#pragma unroll
    for (int i = 0; i < 4; ++i) {
      const int r = lr + 8 * i;
      ra[i] = *(const float4*)(A + (size_t)r * K + k0 + lc);
      rw[i] = *(const float4*)(W + (size_t)(n0 + r) * K + k0 + lc);
    }
#pragma unroll
    for (int i = 0; i < 4; ++i) {
      const int r = lr + 8 * i;
      *(float4*)(&As[0][r * GEMM_LDS_STRIDE + lc]) = ra[i];
      *(float4*)(&Bs[0][r * GEMM_LDS_STRIDE + lc]) = rw[i];
    }
    __syncthreads();
#pragma unroll
    for (int kk = 0; kk < 64; kk += 4) {
      const v2f a = *(const v2f*)(&As[0][(mwarp + lrow) * GEMM_LDS_STRIDE + kk + khalf]);
      const v2f b = *(const v2f*)(&Bs[0][(nwarp + lrow) * GEMM_LDS_STRIDE + kk + khalf]);
      acc = __builtin_amdgcn_wmma_f32_16x16x4_f32(
          false, a, false, b, (short)0, acc, false, false);
    }
    __syncthreads();
  }
#endif

  // C/D layout: VGPR j, lanes 0-15 -> M=j, lanes 16-31 -> M=j+8; N = lane%16.
  const int cm = mwarp + 8 * (lane >> 4);
  const int cn = n0 + nwarp + lrow;
#pragma unroll
  for (int j = 0; j < 8; ++j) {
    C[(size_t)(cm + j) * ldc + cn] = acc[j];
  }
}

// ---------------------------------------------------------------------------
// RoPE on q (32x32x128) and k (32x8x128) in workspace, in place.
// ---------------------------------------------------------------------------
__global__ __launch_bounds__(256) void rope_kernel(
    float* __restrict__ q, float* __restrict__ k,
    const float* __restrict__ cs, const float* __restrict__ sn)
{
  const int idx = blockIdx.x * blockDim.x + threadIdx.x;
  const int total_q = kB * kHQ * (kHD / 2);       // 65536 pairs
  const int total   = total_q + kB * kHKV * (kHD / 2);
  if (idx >= total) return;

  float* buf;
  int p;
  if (idx < total_q) { buf = q; p = idx; }
  else               { buf = k; p = idx - total_q; }

  const int i = p & 63;
  const float c = cs[i];
  const float s = sn[i];
  const float xr = buf[2 * p];
  const float xi = buf[2 * p + 1];
  buf[2 * p]     = xr * c - xi * s;
  buf[2 * p + 1] = xr * s + xi * c;
}

// ---------------------------------------------------------------------------
// GQA attention, one workgroup per (kv head g, batch b). 8 waves stream K/V
// rows as coalesced float4 (512B rows); wave32 xor-shuffle dot reductions;
// softmax over T=2048 in LDS. t == start_pos uses the fresh workspace k/v
// (input caches are never modified).
// ---------------------------------------------------------------------------
__global__ __launch_bounds__(256) void attn_kernel(
    const float* __restrict__ q,       // [32][32][128] (roped)
    const float* __restrict__ knew,    // [32][8][128]  (roped)
    const float* __restrict__ vnew,    // [32][8][128]
    const float* __restrict__ kcache,  // [32][2048][8][128]
    const float* __restrict__ vcache,  // [32][2048][8][128]
    const int* __restrict__ start_pos_p,
    float* __restrict__ out)           // [32][32*128]
{
  const int g    = blockIdx.x;
  const int b    = blockIdx.y;
  const int tid  = threadIdx.x;
  const int wave = tid >> 5;
  const int lane = tid & 31;

  const int start = *start_pos_p;      // 2047
  const int T     = start + 1;         // 2048

  __shared__ float qs[4][kHD];
  __shared__ float sc[4][kMAXS];
  __shared__ float red[256];
  __shared__ float pv[8][4][kHD];

  for (int i = tid; i < 4 * kHD; i += 256) {
    const int r = i >> 7, d = i & 127;
    qs[r][d] = q[((size_t)(b * kHQ + g * 4 + r) << 7) + d];
  }
  __syncthreads();

  const float scale = 0.08838834764831845f;  // 1/sqrt(128)

  // ---- scores ----
  for (int t = wave; t < T; t += 8) {
    const float* krow = (t == start)
        ? (knew + ((size_t)(b * kHKV + g) << 7))
        : (kcache + ((((size_t)b * kMAXS + t) * kHKV + g) << 7));
    const float4 kv = *(const float4*)(krow + lane * 4);
    const float4 q0 = *(const float4*)(&qs[0][lane * 4]);
    const float4 q1 = *(const float4*)(&qs[1][lane * 4]);
    const float4 q2 = *(const float4*)(&qs[2][lane * 4]);
    const float4 q3 = *(const float4*)(&qs[3][lane * 4]);
    float s0 = q0.x * kv.x + q0.y * kv.y + q0.z * kv.z + q0.w * kv.w;
    float s1 = q1.x * kv.x + q1.y * kv.y + q1.z * kv.z + q1.w * kv.w;
    float s2 = q2.x * kv.x + q2.y * kv.y + q2.z * kv.z + q2.w * kv.w;
    float s3 = q3.x * kv.x + q3.y * kv.y + q3.z * kv.z + q3.w * kv.w;
#pragma unroll
    for (int off = 16; off > 0; off >>= 1) {
      s0 += __shfl_xor(s0, off, 32);
      s1 += __shfl_xor(s1, off, 32);
      s2 += __shfl_xor(s2, off, 32);
      s3 += __shfl_xor(s3, off, 32);
    }
    if (lane == 0) {
      sc[0][t] = s0 * scale;
      sc[1][t] = s1 * scale;
      sc[2][t] = s2 * scale;
      sc[3][t] = s3 * scale;
    }
  }
  __syncthreads();

  // ---- softmax ----
  float denom[4];
#pragma unroll
  for (int r = 0; r < 4; ++r) {
    float m = -INFINITY;
    for (int t = tid; t < T; t += 256) m = fmaxf(m, sc[r][t]);
    red[tid] = m;
    __syncthreads();
    for (int s2 = 128; s2 > 0; s2 >>= 1) {
      if (tid < s2) red[tid] = fmaxf(red[tid], red[tid + s2]);
      __syncthreads();
    }
    m = red[0];
    __syncthreads();

    float partial = 0.f;
    for (int t = tid; t < T; t += 256) {
      const float e = __expf(sc[r][t] - m);
      sc[r][t] = e;
      partial += e;
    }
    red[tid] = partial;
    __syncthreads();
    for (int s2 = 128; s2 > 0; s2 >>= 1) {
      if (tid < s2) red[tid] += red[tid + s2];
      __syncthreads();
    }
    denom[r] = red[0];
    __syncthreads();
  }

  // ---- probs @ V ----
  float acc[4][4];
#pragma unroll
  for (int r = 0; r < 4; ++r)
#pragma unroll
    for (int j = 0; j < 4; ++j) acc[r][j] = 0.f;

  for (int t = wave; t < T; t += 8) {
    const float* vrow = (t == start)
        ? (vnew + ((size_t)(b * kHKV + g) << 7))
        : (vcache + ((((size_t)b * kMAXS + t) * kHKV + g) << 7));
    const float4 vv = *(const float4*)(vrow + lane * 4);
#pragma unroll
    for (int r = 0; r < 4; ++r) {
      const float p = sc[r][t];
      acc[r][0] += p * vv.x;
      acc[r][1] += p * vv.y;
      acc[r][2] += p * vv.z;
      acc[r][3] += p * vv.w;
    }
  }
#pragma unroll
  for (int r = 0; r < 4; ++r)
#pragma unroll
    for (int j = 0; j < 4; ++j) pv[wave][r][lane * 4 + j] = acc[r][j];
  __syncthreads();

  for (int i = tid; i < 4 * kHD; i += 256) {
    const int r = i >> 7, d = i & 127;
    float s = 0.f;
#pragma unroll
    for (int w = 0; w < 8; ++w) s += pv[w][r][d];
    out[((size_t)(b * kHQ + g * 4 + r) << 7) + d] = s / denom[r];
  }
}

// ---------------------------------------------------------------------------
// Launcher
// ---------------------------------------------------------------------------
extern "C" void kernel_launch(void* const* d_in, const int* in_sizes, int n_in,
                              void* d_out, int out_size, void* d_ws, size_t ws_size,
                              hipStream_t stream) {
  const float* x   = (const float*)d_in[0];
  const float* fc  = (const float*)d_in[1];
  const float* fs  = (const float*)d_in[2];
  const float* wq  = (const float*)d_in[3];
  const float* wk  = (const float*)d_in[4];
  const float* wv  = (const float*)d_in[5];
  const float* wo  = (const float*)d_in[6];
  const float* kc  = (const float*)d_in[7];
  const float* vc  = (const float*)d_in[8];
  const int*   sp  = (const int*)d_in[9];
  float* out = (float*)d_out;

  float* ws = (float*)d_ws;
  float* q_ws  = ws;                                // 32*4096
  float* k_ws  = q_ws + (size_t)kB * kDIM;          // 32*1024
  float* v_ws  = k_ws + (size_t)kB * kHKV * kHD;    // 32*1024
  float* ao_ws = v_ws + (size_t)kB * kHKV * kHD;    // 32*4096

  gemm_xwT_kernel<<<dim3(kDIM / 32), 128, 0, stream>>>(x, wq, q_ws, kDIM, kDIM, kDIM);
  gemm_xwT_kernel<<<dim3((kHKV * kHD) / 32), 128, 0, stream>>>(x, wk, k_ws, kHKV * kHD, kDIM, kHKV * kHD);
  gemm_xwT_kernel<<<dim3((kHKV * kHD) / 32), 128, 0, stream>>>(x, wv, v_ws, kHKV * kHD, kDIM, kHKV * kHD);

  {
    const int pairs = kB * kHQ * (kHD / 2) + kB * kHKV * (kHD / 2);
    rope_kernel<<<dim3((pairs + 255) / 256), 256, 0, stream>>>(q_ws, k_ws, fc, fs);
  }

  attn_kernel<<<dim3(kHKV, kB), 256, 0, stream>>>(q_ws, k_ws, v_ws, kc, vc, sp, ao_ws);

  gemm_xwT_kernel<<<dim3(kDIM / 32), 128, 0, stream>>>(ao_ws, wo, out, kDIM, kDIM, kDIM);
}